// GAT_40999757807926
// MI455X (gfx1250) — compile-verified
//
#include <hip/hip_runtime.h>
#include <math.h>

typedef __attribute__((ext_vector_type(2))) float v2f;
typedef __attribute__((ext_vector_type(8))) float v8f;

#define NEG_SLOPE 0.2f
#define GAT_EPS 1e-16f

// ---------------------------------------------------------------------------
// Float atomic max via sign-aware integer atomics (monotone bit encoding).
// ---------------------------------------------------------------------------
__device__ __forceinline__ void atomic_max_f32(float* addr, float val) {
  if (val >= 0.0f) {
    atomicMax((int*)addr, __float_as_int(val));
  } else {
    atomicMin((unsigned int*)addr, __float_as_uint(val));
  }
}

// ---------------------------------------------------------------------------
// Fill kernel (init accumulators / -inf for segment max / zero d_out).
// ---------------------------------------------------------------------------
__global__ void gat_fill_f32(float* __restrict__ p, long long n, float v) {
  long long i = blockIdx.x * (long long)blockDim.x + threadIdx.x;
  if (i < n) p[i] = v;
}

// Pad/copy B: Bpad[K, Npad] <- B[K, Nc] (zero beyond Nc).
__global__ void gat_pad_b(const float* __restrict__ B, float* __restrict__ Bp,
                          int K, int Nc, int Npad) {
  const int t = blockIdx.x * blockDim.x + threadIdx.x;
  if (t >= K * Npad) return;
  const int k = t / Npad;
  const int j = t - k * Npad;
  Bp[t] = (j < Nc) ? B[k * Nc + j] : 0.0f;
}

// ---------------------------------------------------------------------------
// FP32 WMMA GEMM: C[M,NPAD] = A[M,K] @ B[K,NPAD], row-major, guard-free.
// K and NPAD are compile-time so the K-loop fully unrolls: every load is
// base + immediate offset (max B offset 128*64*4 = 32KB < 24-bit IOFFSET),
// no remainder loop, no per-step pointer arithmetic.
// One wave per 16x16 output tile, V_WMMA_F32_16X16X4_F32 per 4-wide K step.
//   A fragment (16x4 f32): lane L -> row = L&15; lanes 0-15 hold K=k,k+1,
//     lanes 16-31 hold K=k+2,k+3  -> one aligned b64 load per step.
//   B fragment (4x16 f32): lane L -> col = L&15; same K split.
//   C/D (8 VGPRs): VGPR r -> row r (lanes 0-15) / r+8 (lanes 16-31).
// EXEC is all-ones at every WMMA (ISA requirement): zero divergence.
// ---------------------------------------------------------------------------
template <int K, int NPAD>
__global__ void gat_wmma_gemm_f32(const float* __restrict__ A,
                                  const float* __restrict__ B,
                                  float* __restrict__ C,
                                  int M, int tilesN) {
  const int wave = (int)((blockIdx.x * (long long)blockDim.x + threadIdx.x) >> 5);
  const int lane = threadIdx.x & 31;
  const int tilesM = M >> 4;
  const int total = tilesM * tilesN;
  if (wave >= total) return;  // wave-uniform exit

  const int tm = wave / tilesN;
  const int tn = wave - tm * tilesN;

  const int row   = tm * 16 + (lane & 15);
  const int col   = tn * 16 + (lane & 15);
  const int khalf = (lane >> 4) << 1;  // 0 for lanes 0-15, 2 for lanes 16-31

  const float* Ap = A + (long long)row * K + khalf;
  const float* Bp = B + (long long)khalf * NPAD + col;

  v8f c = {};
#pragma unroll
  for (int k = 0; k < K; k += 4) {
    const v2f a = *(const v2f*)(Ap + k);  // aligned b64, immediate offset
    v2f b;
    b.x = Bp[k * NPAD];
    b.y = Bp[(k + 1) * NPAD];
    c = __builtin_amdgcn_wmma_f32_16x16x4_f32(
        /*neg_a=*/false, a, /*neg_b=*/false, b,
        /*c_mod=*/(short)0, c, /*reuse_a=*/false, /*reuse_b=*/false);
  }

  const int half = lane >> 4;  // 0 -> rows r, 1 -> rows r+8
  float* Cp = C + (long long)(tm * 16 + half * 8) * NPAD + col;
#pragma unroll
  for (int r = 0; r < 8; ++r) {
    Cp[r * NPAD] = c[r];
  }
}

// ---------------------------------------------------------------------------
// alpha_s[n,h] = dot(h[n,h,:], a_src[h,:]);  alpha_d likewise.
// Grid: x over nodes, y over heads.  h row stride = ldh (padded).
// ---------------------------------------------------------------------------
__global__ void gat_alpha(const float* __restrict__ h,
                          const float* __restrict__ a_src,
                          const float* __restrict__ a_dst,
                          float* __restrict__ as, float* __restrict__ ad,
                          int N, int H, int C, int ldh) {
  const int n = blockIdx.x * blockDim.x + threadIdx.x;
  if (n >= N) return;
  const int hh = blockIdx.y;
  const float* hp = h + (long long)n * ldh + (long long)hh * C;
  const float* sp = a_src + (long long)hh * C;
  const float* dp = a_dst + (long long)hh * C;
  float s = 0.0f, d = 0.0f;
  for (int c = 0; c < C; ++c) {
    const float v = hp[c];
    s += v * sp[c];
    d += v * dp[c];
  }
  as[(long long)n * H + hh] = s;
  ad[(long long)n * H + hh] = d;
}

// Decode edge id -> (src, dst); ids in [E, E+N) are self-loops.
__device__ __forceinline__ void gat_edge(const long long* __restrict__ ei,
                                         long long e, long long E,
                                         long long& s, long long& d) {
  if (e < E) {
    s = ei[e];
    d = ei[E + e];
  } else {
    s = d = e - E;
  }
}

__device__ __forceinline__ float gat_e_val(const float* __restrict__ as,
                                           const float* __restrict__ ad,
                                           long long s, long long d,
                                           int H, int h) {
  float v = as[s * H + h] + ad[d * H + h];
  return v > 0.0f ? v : NEG_SLOPE * v;
}

// Pass 1: per-destination segment max of LeakyReLU(e).  Grid y = head.
__global__ void gat_edge_max(const long long* __restrict__ ei, long long E, int N,
                             int H, const float* __restrict__ as,
                             const float* __restrict__ ad, float* __restrict__ m) {
  const long long e = blockIdx.x * (long long)blockDim.x + threadIdx.x;
  if (e >= E + N) return;
  const int h = blockIdx.y;
  long long s, d;
  gat_edge(ei, e, E, s, d);
  atomic_max_f32(&m[d * H + h], gat_e_val(as, ad, s, d, H, h));
}

// Pass 2: segment sum of exp(e - m[dst]).  Grid y = head.
__global__ void gat_edge_sum(const long long* __restrict__ ei, long long E, int N,
                             int H, const float* __restrict__ as,
                             const float* __restrict__ ad,
                             const float* __restrict__ m, float* __restrict__ ssum) {
  const long long e = blockIdx.x * (long long)blockDim.x + threadIdx.x;
  if (e >= E + N) return;
  const int h = blockIdx.y;
  long long s, d;
  gat_edge(ei, e, E, s, d);
  const float v = gat_e_val(as, ad, s, d, H, h);
  atomicAdd(&ssum[d * H + h], expf(v - m[d * H + h]));
}

// Pass 3: out[dst,h,:] += alpha * h[src,h,:], alpha = ex / (sum + eps).
// Grid y = head.  Gather via float4 (all strides 16B-aligned), scatter via
// per-component global_atomic_add_f32.  ldsrc/lddst = row strides (floats).
__global__ void gat_edge_aggr(const long long* __restrict__ ei, long long E, int N,
                              int H, int C, int ldsrc, int lddst,
                              const float* __restrict__ as,
                              const float* __restrict__ ad,
                              const float* __restrict__ m,
                              const float* __restrict__ ssum,
                              const float* __restrict__ hfeat,
                              float* __restrict__ acc) {
  const long long e = blockIdx.x * (long long)blockDim.x + threadIdx.x;
  if (e >= E + N) return;
  const int h = blockIdx.y;
  long long s, d;
  gat_edge(ei, e, E, s, d);
  const float v = gat_e_val(as, ad, s, d, H, h);
  const float alpha = expf(v - m[d * H + h]) / (ssum[d * H + h] + GAT_EPS);
  const float* sp = hfeat + (long long)s * ldsrc + (long long)h * C;
  float* dp = acc + (long long)d * lddst + (long long)h * C;
  for (int c = 0; c < C; c += 4) {
    const float4 f = *(const float4*)(sp + c);
    atomicAdd(&dp[c + 0], f.x * alpha);
    atomicAdd(&dp[c + 1], f.y * alpha);
    atomicAdd(&dp[c + 2], f.z * alpha);
    atomicAdd(&dp[c + 3], f.w * alpha);
  }
}

// In-place: x = ELU(x + bias)  (layer-1 epilogue; output feeds GEMM2).
__global__ void gat_bias_elu(float* __restrict__ x, const float* __restrict__ b,
                             long long total, int F) {
  const long long t = blockIdx.x * (long long)blockDim.x + threadIdx.x;
  if (t >= total) return;
  const int f = (int)(t % F);
  const float v = x[t] + b[f];
  x[t] = v > 0.0f ? v : expm1f(v);
}

// In-place: x += bias  (final output epilogue).
__global__ void gat_bias_add(float* __restrict__ x, const float* __restrict__ b,
                             long long total, int F) {
  const long long t = blockIdx.x * (long long)blockDim.x + threadIdx.x;
  if (t >= total) return;
  x[t] += b[(int)(t % F)];
}

// ---------------------------------------------------------------------------
// Launcher
// ---------------------------------------------------------------------------
extern "C" void kernel_launch(void* const* d_in, const int* in_sizes, int n_in,
                              void* d_out, int out_size, void* d_ws, size_t ws_size,
                              hipStream_t stream) {
  const float*     x      = (const float*)d_in[0];
  const long long* ei     = (const long long*)d_in[1];  // int64 edge_index [2,E]
  const float*     W1     = (const float*)d_in[2];
  const float*     a_src1 = (const float*)d_in[3];
  const float*     a_dst1 = (const float*)d_in[4];
  const float*     b1     = (const float*)d_in[5];
  const float*     W2     = (const float*)d_in[6];
  const float*     a_src2 = (const float*)d_in[7];
  const float*     a_dst2 = (const float*)d_in[8];
  const float*     b2     = (const float*)d_in[9];
  float*           out    = (float*)d_out;

  const int FIN = 128, H1 = 4, C1 = 16, F1 = 64, F2 = 40, F2P = 48;
  const int       N = in_sizes[0] / FIN;            // 50000 (multiple of 16)
  const long long E = (long long)in_sizes[1] / 2;   // 1000000

  // Workspace layout (floats)
  float* ws    = (float*)d_ws;
  float* h1    = ws;                           // [N,64]   x @ W1
  float* acc1  = h1    + (size_t)N * F1;       // [N,64]   layer-1 agg -> ELU'd h
  float* h2    = acc1  + (size_t)N * F1;       // [N,48]   h @ W2 (padded ld)
  float* W2pad = h2    + (size_t)N * F2P;      // [64,48]  zero-padded W2
  float* as1   = W2pad + (size_t)F1 * F2P;     // [N,4]
  float* ad1   = as1   + (size_t)N * H1;       // [N,4]
  float* m1    = ad1   + (size_t)N * H1;       // [N,4]
  float* s1    = m1    + (size_t)N * H1;       // [N,4]
  float* as2   = s1    + (size_t)N * H1;       // [N]
  float* ad2   = as2   + (size_t)N;            // [N]
  float* m2    = ad2   + (size_t)N;            // [N]
  float* s2    = m2    + (size_t)N;            // [N]

  auto cdiv = [](long long a, long long b) { return (int)((a + b - 1) / b); };
  const float NEG_INF = -__builtin_inff();

  // ---- init accumulators ----
  gat_fill_f32<<<cdiv((long long)N * F1, 256), 256, 0, stream>>>(acc1, (long long)N * F1, 0.0f);
  gat_fill_f32<<<cdiv((long long)N * H1, 256), 256, 0, stream>>>(m1, (long long)N * H1, NEG_INF);
  gat_fill_f32<<<cdiv((long long)N * H1, 256), 256, 0, stream>>>(s1, (long long)N * H1, 0.0f);
  gat_fill_f32<<<cdiv(N, 256), 256, 0, stream>>>(m2, N, NEG_INF);
  gat_fill_f32<<<cdiv(N, 256), 256, 0, stream>>>(s2, N, 0.0f);
  gat_fill_f32<<<cdiv((long long)N * F2, 256), 256, 0, stream>>>(out, (long long)N * F2, 0.0f);
  gat_pad_b<<<cdiv(F1 * F2P, 256), 256, 0, stream>>>(W2, W2pad, F1, F2, F2P);

  // ---- layer 1 ----
  const int tilesM  = N / 16;           // 3125
  const int tilesN1 = F1 / 16;          // 4
  const int waves1  = tilesM * tilesN1;
  gat_wmma_gemm_f32<128, 64><<<cdiv(waves1, 4), 128, 0, stream>>>(x, W1, h1, N, tilesN1);

  gat_alpha<<<dim3(cdiv(N, 256), H1), 256, 0, stream>>>(h1, a_src1, a_dst1, as1, ad1, N, H1, C1, F1);

  const int eblk = cdiv(E + N, 256);
  gat_edge_max <<<dim3(eblk, H1), 256, 0, stream>>>(ei, E, N, H1, as1, ad1, m1);
  gat_edge_sum <<<dim3(eblk, H1), 256, 0, stream>>>(ei, E, N, H1, as1, ad1, m1, s1);
  gat_edge_aggr<<<dim3(eblk, H1), 256, 0, stream>>>(ei, E, N, H1, C1, F1, F1,
                                                    as1, ad1, m1, s1, h1, acc1);
  gat_bias_elu <<<cdiv((long long)N * F1, 256), 256, 0, stream>>>(acc1, b1, (long long)N * F1, F1);

  // ---- layer 2 ----
  const int tilesN2 = F2P / 16;         // 3
  const int waves2  = tilesM * tilesN2;
  gat_wmma_gemm_f32<64, 48><<<cdiv(waves2, 4), 128, 0, stream>>>(acc1, W2pad, h2, N, tilesN2);

  gat_alpha<<<dim3(cdiv(N, 256), 1), 256, 0, stream>>>(h2, a_src2, a_dst2, as2, ad2, N, 1, F2, F2P);

  gat_edge_max <<<dim3(eblk, 1), 256, 0, stream>>>(ei, E, N, 1, as2, ad2, m2);
  gat_edge_sum <<<dim3(eblk, 1), 256, 0, stream>>>(ei, E, N, 1, as2, ad2, m2, s2);
  gat_edge_aggr<<<dim3(eblk, 1), 256, 0, stream>>>(ei, E, N, 1, F2, F2P, F2,
                                                   as2, ad2, m2, s2, h2, out);
  gat_bias_add <<<cdiv((long long)N * F2, 256), 256, 0, stream>>>(out, b2, (long long)N * F2, F2);
}